// TransformerBlock_55078660604238
// MI455X (gfx1250) — compile-verified
//
#include <hip/hip_runtime.h>

// ---------------- CDNA5 WMMA f32 16x16x4 helpers ----------------
typedef __attribute__((ext_vector_type(2))) float v2f;
typedef __attribute__((ext_vector_type(8))) float v8f;

#define LDT 66   // padded LDS activation row stride (floats)
#define WSK 80   // LDS weight stride per k2 row, in v2f (skewed: 160 dw % 64 = 32)

__device__ __forceinline__ v8f wmma4(v2f a, v2f b, v8f c) {
  // D = A(16x4 f32) x B(4x16 f32) + C(16x16 f32)
  return __builtin_amdgcn_wmma_f32_16x16x4_f32(false, a, false, b, (short)0, c,
                                               false, false);
}

__device__ __forceinline__ void zero_acc(v8f acc[4]) {
#pragma unroll
  for (int t = 0; t < 4; ++t)
#pragma unroll
    for (int r = 0; r < 8; ++r) acc[t][r] = 0.0f;
}

// Stage a 64x64 row-major global weight matrix into LDS, k-paired + skewed:
// Wl2[k2*WSK + col] = { W[2*k2][col], W[2*k2+1][col] }   (k2 = 0..31)
__device__ __forceinline__ void stage_W(const float* __restrict__ W,
                                        v2f* __restrict__ Wl2) {
  for (int i = threadIdx.x; i < 32 * 64; i += 256) {
    const int k2 = i >> 6, col = i & 63;
    v2f p;
    p.x = W[(2 * k2) * 64 + col];
    p.y = W[(2 * k2 + 1) * 64 + col];
    Wl2[k2 * WSK + col] = p;
  }
}

// D[16x64] += A_lds[16x64 (stride LDT)] @ W (k-paired LDS layout)
__device__ __forceinline__ void gemm_ll(const float* __restrict__ A,
                                        const v2f* __restrict__ Wl2,
                                        v8f acc[4]) {
  const int lane = threadIdx.x & 31;
  const int row  = lane & 15;
  const int hi   = (lane >> 4) & 1;   // lanes 16-31 carry K+2,K+3
  const int nc   = lane & 15;
#pragma unroll
  for (int k0 = 0; k0 < 64; k0 += 4) {
    const int k2 = (k0 >> 1) + hi;
    const v2f a = *(const v2f*)&A[row * LDT + k0 + 2 * hi];
#pragma unroll
    for (int nt = 0; nt < 4; ++nt) {
      const v2f b = Wl2[k2 * WSK + nt * 16 + nc];
      acc[nt] = wmma4(a, b, acc[nt]);
    }
  }
}

// store D fragments (16x64) to an LDS tile, optional bias + relu
__device__ __forceinline__ void store_lds(float* __restrict__ Dst,
                                          const v8f acc[4],
                                          const float* __restrict__ bias,
                                          bool relu) {
  const int lane = threadIdx.x & 31;
  const int hi = (lane >> 4) & 1;     // C/D: vgpr r -> M = r + 8*hi
  const int nc = lane & 15;
#pragma unroll
  for (int nt = 0; nt < 4; ++nt) {
    const int col = nt * 16 + nc;
    const float bv = bias ? bias[col] : 0.0f;
#pragma unroll
    for (int r = 0; r < 8; ++r) {
      float x = acc[nt][r] + bv;
      if (relu) x = fmaxf(x, 0.0f);
      Dst[(r + 8 * hi) * LDT + col] = x;
    }
  }
}

// store D fragments to global rows [base .. base+15] x 64, with row guard
__device__ __forceinline__ void store_global_rows(float* __restrict__ out,
                                                  int base, int nrows,
                                                  const v8f acc[4],
                                                  const float* __restrict__ bias,
                                                  bool relu) {
  const int lane = threadIdx.x & 31;
  const int hi = (lane >> 4) & 1;
  const int nc = lane & 15;
#pragma unroll
  for (int nt = 0; nt < 4; ++nt) {
    const int col = nt * 16 + nc;
    const float bv = bias ? bias[col] : 0.0f;
#pragma unroll
    for (int r = 0; r < 8; ++r) {
      const int m = base + r + 8 * hi;
      if (m < nrows) {
        float x = acc[nt][r] + bv;
        if (relu) x = fmaxf(x, 0.0f);
        out[(size_t)m * 64 + col] = x;
      }
    }
  }
}

// ---------------- Pass A: node projections ----------------
// h = relu(x@W_in + b_in); a_dst = h@W_dst; a_src = h@W_src; v = h@W_lin
__global__ __launch_bounds__(256) void node_proj_kernel(
    const float* __restrict__ x, const float* __restrict__ W_in,
    const float* __restrict__ b_in, const float* __restrict__ W_dst,
    const float* __restrict__ W_src, const float* __restrict__ W_lin,
    float* __restrict__ a_dst, float* __restrict__ a_src,
    float* __restrict__ v, int N) {
  __shared__ __align__(16) float sh_h[8][16 * LDT];
  __shared__ v2f sh_w[4][32 * WSK];
  const int w = threadIdx.x >> 5;
  const int lane = threadIdx.x & 31;
  const int nb = (blockIdx.x * 8 + w) * 16;

  stage_W(W_in,  sh_w[0]);
  stage_W(W_dst, sh_w[1]);
  stage_W(W_src, sh_w[2]);
  stage_W(W_lin, sh_w[3]);
  __syncthreads();
  if (nb >= N) return;  // wave-uniform

  const int row = lane & 15;
  const int hi = (lane >> 4) & 1;
  const int nc = lane & 15;

  // stage 1: h tile, A sourced straight from global x (clamped rows)
  v8f acc[4];
  zero_acc(acc);
  int gr = nb + row;
  if (gr >= N) gr = N - 1;
  const float* xrow = x + (size_t)gr * 64;
#pragma unroll
  for (int k0 = 0; k0 < 64; k0 += 4) {
    const int k2 = (k0 >> 1) + hi;
    const v2f a = *(const v2f*)&xrow[k0 + 2 * hi];
#pragma unroll
    for (int nt = 0; nt < 4; ++nt) {
      const v2f b = sh_w[0][k2 * WSK + nt * 16 + nc];
      acc[nt] = wmma4(a, b, acc[nt]);
    }
  }
  store_lds(sh_h[w], acc, b_in, /*relu=*/true);

  // stage 2: three projections off the h tile (bias=False, no relu)
  zero_acc(acc);
  gemm_ll(sh_h[w], sh_w[1], acc);
  store_global_rows(a_dst, nb, N, acc, nullptr, false);

  zero_acc(acc);
  gemm_ll(sh_h[w], sh_w[2], acc);
  store_global_rows(a_src, nb, N, acc, nullptr, false);

  zero_acc(acc);
  gemm_ll(sh_h[w], sh_w[3], acc);
  store_global_rows(v, nb, N, acc, nullptr, false);
}

// ---------------- shared edge-tile MLP pipeline ----------------
// Produces: t1 = attn logits a (16xLDT), dl = delta (16xLDT), s_src/s_dst idx
__device__ __forceinline__ void edge_tile_compute(
    int eb, int E, const float* __restrict__ pos, const int* __restrict__ srcI,
    const int* __restrict__ dstI, const float* __restrict__ Wp1,
    const float* __restrict__ bp1, const v2f* __restrict__ Wp2l,
    const float* __restrict__ bp2, const v2f* __restrict__ Wa1l,
    const float* __restrict__ ba1, const v2f* __restrict__ Wa2l,
    const float* __restrict__ ba2, const float* __restrict__ a_dst,
    const float* __restrict__ a_src, float* __restrict__ t1,
    float* __restrict__ dl, int* __restrict__ s_src, int* __restrict__ s_dst) {
  const int lane = threadIdx.x & 31;
  const int row = lane & 15;
  const int hi = (lane >> 4) & 1;
  const int nc = lane & 15;

  // fetch edge endpoints for this 16-edge tile
  {
    int e = eb + row;
    if (e >= E) e = E - 1;
    if (hi == 0) s_src[row] = srcI[e];
    else         s_dst[row] = dstI[e];
  }

  // pos_nn layer 1: dpos(16x2, K padded to 4) @ Wp1(2x64) + bp1, relu
  v8f acc[4];
  zero_acc(acc);
  v2f a;
  if (hi == 0) {
    const int d = s_dst[row], s = s_src[row];
    a.x = pos[(size_t)d * 2 + 0] - pos[(size_t)s * 2 + 0];
    a.y = pos[(size_t)d * 2 + 1] - pos[(size_t)s * 2 + 1];
  } else {
    a.x = 0.0f; a.y = 0.0f;    // K=2,3 are zero-pad
  }
#pragma unroll
  for (int nt = 0; nt < 4; ++nt) {
    const int col = nt * 16 + nc;
    v2f b;
    if (hi == 0) { b.x = Wp1[0 * 64 + col]; b.y = Wp1[1 * 64 + col]; }
    else         { b.x = 0.0f; b.y = 0.0f; }
    acc[nt] = wmma4(a, b, acc[nt]);
  }
  store_lds(t1, acc, bp1, true);

  // pos_nn layer 2: delta = relu(t1 @ Wp2 + bp2)
  zero_acc(acc);
  gemm_ll(t1, Wp2l, acc);
  store_lds(dl, acc, bp2, true);

  // araw = a_dst[dst] - a_src[src] + delta  (cooperative gather into t1)
#pragma unroll 4
  for (int i = 0; i < 32; ++i) {
    const int flat = i * 32 + lane;
    const int e = flat >> 6, c = flat & 63;
    const int d = s_dst[e], s = s_src[e];
    t1[e * LDT + c] =
        a_dst[(size_t)d * 64 + c] - a_src[(size_t)s * 64 + c] + dl[e * LDT + c];
  }

  // attn_nn: two 64x64 layers with relu, in-place on t1
  zero_acc(acc);
  gemm_ll(t1, Wa1l, acc);
  store_lds(t1, acc, ba1, true);

  zero_acc(acc);
  gemm_ll(t1, Wa2l, acc);
  store_lds(t1, acc, ba2, true);
}

// ---------------- Pass B: segment max (atomicMax, a >= 0) ----------------
__global__ __launch_bounds__(256) void edge_max_kernel(
    const float* __restrict__ pos, const int* __restrict__ srcI,
    const int* __restrict__ dstI, const float* __restrict__ Wp1,
    const float* __restrict__ bp1, const float* __restrict__ Wp2,
    const float* __restrict__ bp2, const float* __restrict__ Wa1,
    const float* __restrict__ ba1, const float* __restrict__ Wa2,
    const float* __restrict__ ba2, const float* __restrict__ a_dst,
    const float* __restrict__ a_src, int* __restrict__ amax_i, int E) {
  __shared__ __align__(16) float sh_t1[8][16 * LDT];
  __shared__ __align__(16) float sh_dl[8][16 * LDT];
  __shared__ int sh_s[8][16];
  __shared__ int sh_d[8][16];
  __shared__ v2f sh_w[3][32 * WSK];
  const int w = threadIdx.x >> 5;
  const int lane = threadIdx.x & 31;
  const int eb = (blockIdx.x * 8 + w) * 16;

  stage_W(Wp2, sh_w[0]);
  stage_W(Wa1, sh_w[1]);
  stage_W(Wa2, sh_w[2]);
  __syncthreads();
  if (eb >= E) return;

  edge_tile_compute(eb, E, pos, srcI, dstI, Wp1, bp1, sh_w[0], bp2, sh_w[1],
                    ba1, sh_w[2], ba2, a_dst, a_src, sh_t1[w], sh_dl[w],
                    sh_s[w], sh_d[w]);

#pragma unroll 4
  for (int i = 0; i < 32; ++i) {
    const int flat = i * 32 + lane;
    const int e = flat >> 6, c = flat & 63;
    if (eb + e < E) {
      const float val = sh_t1[w][e * LDT + c];  // >= 0 (post-relu)
      atomicMax(&amax_i[(size_t)sh_d[w][e] * 64 + c], __float_as_int(val));
    }
  }
}

// ---------------- Pass C: exp + fused numerator/denominator sums ----------------
__global__ __launch_bounds__(256) void edge_sum_kernel(
    const float* __restrict__ pos, const int* __restrict__ srcI,
    const int* __restrict__ dstI, const float* __restrict__ Wp1,
    const float* __restrict__ bp1, const float* __restrict__ Wp2,
    const float* __restrict__ bp2, const float* __restrict__ Wa1,
    const float* __restrict__ ba1, const float* __restrict__ Wa2,
    const float* __restrict__ ba2, const float* __restrict__ a_dst,
    const float* __restrict__ a_src, const float* __restrict__ vbuf,
    const float* __restrict__ amax, float* __restrict__ denom,
    float* __restrict__ numer, int E) {
  __shared__ __align__(16) float sh_t1[8][16 * LDT];
  __shared__ __align__(16) float sh_dl[8][16 * LDT];
  __shared__ int sh_s[8][16];
  __shared__ int sh_d[8][16];
  __shared__ v2f sh_w[3][32 * WSK];
  const int w = threadIdx.x >> 5;
  const int lane = threadIdx.x & 31;
  const int eb = (blockIdx.x * 8 + w) * 16;

  stage_W(Wp2, sh_w[0]);
  stage_W(Wa1, sh_w[1]);
  stage_W(Wa2, sh_w[2]);
  __syncthreads();
  if (eb >= E) return;

  edge_tile_compute(eb, E, pos, srcI, dstI, Wp1, bp1, sh_w[0], bp2, sh_w[1],
                    ba1, sh_w[2], ba2, a_dst, a_src, sh_t1[w], sh_dl[w],
                    sh_s[w], sh_d[w]);

#pragma unroll 4
  for (int i = 0; i < 32; ++i) {
    const int flat = i * 32 + lane;
    const int e = flat >> 6, c = flat & 63;
    if (eb + e < E) {
      const int d = sh_d[w][e], s = sh_s[w][e];
      const float m = amax[(size_t)d * 64 + c];
      const float ex = __expf(sh_t1[w][e * LDT + c] - m);
      atomicAdd(&denom[(size_t)d * 64 + c], ex);
      const float val = vbuf[(size_t)s * 64 + c] + sh_dl[w][e * LDT + c];
      atomicAdd(&numer[(size_t)d * 64 + c], ex * val);
    }
  }
}

// ---------------- Pass D: out = relu((numer/(denom+eps)) @ W_out + b_out) ----------------
__global__ __launch_bounds__(256) void out_kernel(
    const float* __restrict__ numer, const float* __restrict__ denom,
    const float* __restrict__ W_out, const float* __restrict__ b_out,
    float* __restrict__ out, int N) {
  __shared__ __align__(16) float sh_r[8][16 * LDT];
  __shared__ v2f sh_w[32 * WSK];
  const int w = threadIdx.x >> 5;
  const int lane = threadIdx.x & 31;
  const int nb = (blockIdx.x * 8 + w) * 16;

  stage_W(W_out, sh_w);
  __syncthreads();
  if (nb >= N) return;

#pragma unroll 4
  for (int i = 0; i < 32; ++i) {
    const int flat = i * 32 + lane;
    const int r = flat >> 6, c = flat & 63;
    int g = nb + r;
    if (g >= N) g = N - 1;
    sh_r[w][r * LDT + c] =
        numer[(size_t)g * 64 + c] / (denom[(size_t)g * 64 + c] + 1e-16f);
  }

  v8f acc[4];
  zero_acc(acc);
  gemm_ll(sh_r[w], sh_w, acc);
  store_global_rows(out, nb, N, acc, b_out, /*relu=*/true);
}

// ---------------- launcher ----------------
extern "C" void kernel_launch(void* const* d_in, const int* in_sizes, int n_in,
                              void* d_out, int out_size, void* d_ws,
                              size_t ws_size, hipStream_t stream) {
  const float* x     = (const float*)d_in[0];
  const float* pos   = (const float*)d_in[1];
  const int*   eidx  = (const int*)d_in[2];
  const float* W_in  = (const float*)d_in[3];
  const float* b_in  = (const float*)d_in[4];
  const float* W_lin = (const float*)d_in[5];
  const float* W_src = (const float*)d_in[6];
  const float* W_dst = (const float*)d_in[7];
  const float* Wp1   = (const float*)d_in[8];
  const float* bp1   = (const float*)d_in[9];
  const float* Wp2   = (const float*)d_in[10];
  const float* bp2   = (const float*)d_in[11];
  const float* Wa1   = (const float*)d_in[12];
  const float* ba1   = (const float*)d_in[13];
  const float* Wa2   = (const float*)d_in[14];
  const float* ba2   = (const float*)d_in[15];
  const float* W_out = (const float*)d_in[16];
  const float* b_out = (const float*)d_in[17];

  const int N = in_sizes[0] / 64;
  const int E = in_sizes[2] / 2;
  const int* srcI = eidx;       // edge_index[0] = j (message source)
  const int* dstI = eidx + E;   // edge_index[1] = i (message target)

  float* ws    = (float*)d_ws;
  const size_t NC = (size_t)N * 64;
  float* a_dst = ws;            // [N,64]
  float* a_src = a_dst + NC;    // [N,64]
  float* vbuf  = a_src + NC;    // [N,64]
  float* amax  = vbuf + NC;     // [N,64] (bit pattern 0 == 0.0f)
  float* denom = amax + NC;     // [N,64]
  float* numer = denom + NC;    // [N,64]

  // zero amax/denom/numer every call (deterministic, capture-safe)
  hipMemsetAsync(amax, 0, 3 * NC * sizeof(float), stream);

  const int node_blocks = (N + 127) / 128;  // 8 waves x 16 rows per block
  const int edge_blocks = (E + 127) / 128;  // 8 waves x 16 edges per block

  node_proj_kernel<<<node_blocks, 256, 0, stream>>>(
      x, W_in, b_in, W_dst, W_src, W_lin, a_dst, a_src, vbuf, N);

  edge_max_kernel<<<edge_blocks, 256, 0, stream>>>(
      pos, srcI, dstI, Wp1, bp1, Wp2, bp2, Wa1, ba1, Wa2, ba2, a_dst, a_src,
      (int*)amax, E);

  edge_sum_kernel<<<edge_blocks, 256, 0, stream>>>(
      pos, srcI, dstI, Wp1, bp1, Wp2, bp2, Wa1, ba1, Wa2, ba2, a_dst, a_src,
      vbuf, amax, denom, numer, E);

  out_kernel<<<node_blocks, 256, 0, stream>>>(numer, denom, W_out, b_out,
                                              (float*)d_out, N);
}